// MHA_90048284328136
// MI455X (gfx1250) — compile-verified
//
#include <hip/hip_runtime.h>
#include <math.h>

// ---------------------------------------------------------------------------
// MHA forward for MI455X (gfx1250), fp32 via V_WMMA_F32_16X16X4_F32.
// B=4, S=1024, DIM=1024, H=16, Dh=64. wave32.
// Round 2: software-pipelined GEMM (register double-buffering, 32x64 tiles),
// batched fragment loads in attention, global_prefetch for next KV chunk.
// ---------------------------------------------------------------------------

typedef __attribute__((ext_vector_type(2))) float v2f;
typedef __attribute__((ext_vector_type(8))) float v8f;

#define BB   4
#define SS   1024
#define DIMM 1024
#define NH   16
#define DH   64
#define MTOT (BB * SS)          // 4096 rows

__device__ __forceinline__ v8f wmma_f32(v2f a, v2f b, v8f c) {
  // D = A(16x4) x B(4x16) + C(16x16), all fp32.
  return __builtin_amdgcn_wmma_f32_16x16x4_f32(
      /*neg_a=*/false, a, /*neg_b=*/false, b,
      /*c_mod=*/(short)0, c, /*reuse_a=*/false, /*reuse_b=*/false);
}

// Fragment group = 2 k-steps (K advance of 8) for a 32(M)x64(N) wave tile:
// 2 A-frags + 4 B-frags per step = 12 v2f loads feeding 16 WMMAs.
__device__ __forceinline__ void load_grp(const float* a0p, const float* a1p,
                                         const float* w0, const float* w1,
                                         const float* w2, const float* w3,
                                         int kk, v2f g[12]) {
  g[0]  = *(const v2f*)(a0p + kk);
  g[1]  = *(const v2f*)(a1p + kk);
  g[2]  = *(const v2f*)(w0 + kk);
  g[3]  = *(const v2f*)(w1 + kk);
  g[4]  = *(const v2f*)(w2 + kk);
  g[5]  = *(const v2f*)(w3 + kk);
  g[6]  = *(const v2f*)(a0p + kk + 4);
  g[7]  = *(const v2f*)(a1p + kk + 4);
  g[8]  = *(const v2f*)(w0 + kk + 4);
  g[9]  = *(const v2f*)(w1 + kk + 4);
  g[10] = *(const v2f*)(w2 + kk + 4);
  g[11] = *(const v2f*)(w3 + kk + 4);
}

__device__ __forceinline__ void mma_grp(const v2f g[12], v8f acc[8]) {
#pragma unroll
  for (int s = 0; s < 2; ++s) {
    const v2f a0 = g[6 * s + 0];
    const v2f a1 = g[6 * s + 1];
    acc[0] = wmma_f32(a0, g[6 * s + 2], acc[0]);
    acc[1] = wmma_f32(a0, g[6 * s + 3], acc[1]);
    acc[2] = wmma_f32(a0, g[6 * s + 4], acc[2]);
    acc[3] = wmma_f32(a0, g[6 * s + 5], acc[3]);
    acc[4] = wmma_f32(a1, g[6 * s + 2], acc[4]);
    acc[5] = wmma_f32(a1, g[6 * s + 3], acc[5]);
    acc[6] = wmma_f32(a1, g[6 * s + 4], acc[6]);
    acc[7] = wmma_f32(a1, g[6 * s + 5], acc[7]);
  }
}

// ---------------------------------------------------------------------------
// C[M,N] = A[M,K] @ W[N,K]^T (+ bias[N]).  One wave computes 32(M) x 64(N).
// Register double-buffered over K so next-group loads are in flight while the
// current group's 16 WMMAs execute (partial s_wait_loadcnt, not 0-drains).
// ---------------------------------------------------------------------------
__global__ __launch_bounds__(256) void gemm_nt_wmma(
    const float* __restrict__ A, const float* __restrict__ W,
    const float* __restrict__ bias, float* __restrict__ C,
    int M, int N, int K) {
  const int lane = threadIdx.x & 31;
  const int wid  = threadIdx.x >> 5;
  const int half = lane >> 4;   // 0..1
  const int lm   = lane & 15;   // 0..15

  const int ntiles = N >> 6;                 // 64-wide n-tiles
  const int tile = blockIdx.x * 8 + wid;     // 8 waves/block share one m-tile
  const int mt = tile / ntiles;
  const int nt = tile % ntiles;
  const int m0 = mt * 32;
  const int n0 = nt * 64;
  if (m0 >= M) return;

  v8f acc[8];
#pragma unroll
  for (int i = 0; i < 8; ++i) acc[i] = (v8f){};

  const float* a0p = A + (size_t)(m0 + lm) * K + 2 * half;
  const float* a1p = A + (size_t)(m0 + 16 + lm) * K + 2 * half;
  const float* w0  = W + (size_t)(n0 +  0 + lm) * K + 2 * half;
  const float* w1  = W + (size_t)(n0 + 16 + lm) * K + 2 * half;
  const float* w2  = W + (size_t)(n0 + 32 + lm) * K + 2 * half;
  const float* w3  = W + (size_t)(n0 + 48 + lm) * K + 2 * half;

  v2f cur[12], nxt[12];
  load_grp(a0p, a1p, w0, w1, w2, w3, 0, cur);
#pragma unroll 2
  for (int kk = 8; kk < K; kk += 8) {
    load_grp(a0p, a1p, w0, w1, w2, w3, kk, nxt);   // in flight during...
    mma_grp(cur, acc);                             // ...these 16 WMMAs
#pragma unroll
    for (int i = 0; i < 12; ++i) cur[i] = nxt[i];
  }
  mma_grp(cur, acc);

  // Epilogue: acc[ms*4+j][r] = C[m0 + 16*ms + r + 8*half][n0 + 16*j + lm]
#pragma unroll
  for (int ms = 0; ms < 2; ++ms) {
#pragma unroll
    for (int j = 0; j < 4; ++j) {
      const int n = n0 + 16 * j + lm;
      const float bv = bias ? bias[n] : 0.0f;
#pragma unroll
      for (int r = 0; r < 8; ++r) {
        C[(size_t)(m0 + 16 * ms + r + 8 * half) * N + n] = acc[ms * 4 + j][r] + bv;
      }
    }
  }
}

// ---------------------------------------------------------------------------
// RoPE over the FULL last dim (1024) of xq and xk, positions = row % S.
// ---------------------------------------------------------------------------
__global__ __launch_bounds__(256) void rope_kernel(float* __restrict__ xq,
                                                   float* __restrict__ xk) {
  const int idx = blockIdx.x * blockDim.x + threadIdx.x;  // MTOT * DIMM/2
  if (idx >= MTOT * (DIMM / 2)) return;
  const int row = idx >> 9;       // DIMM/2 = 512
  const int i   = idx & 511;
  const int pos = row & (SS - 1); // row = b*S + s
  const float freq = __expf(-(float)(2 * i) * (9.210340371976184f / 1024.0f));
  const float ang = (float)pos * freq;
  float sn, cs;
  __sincosf(ang, &sn, &cs);
  const size_t o = (size_t)row * DIMM + 2 * i;
  const float q0 = xq[o], q1 = xq[o + 1];
  xq[o]     = q0 * cs - q1 * sn;
  xq[o + 1] = q0 * sn + q1 * cs;
  const float k0 = xk[o], k1 = xk[o + 1];
  xk[o]     = k0 * cs - k1 * sn;
  xk[o + 1] = k0 * sn + k1 * cs;
}

// ---------------------------------------------------------------------------
// Flash-style attention (no causal mask). One wave per (b, h, 16-row q-tile),
// streaming 64 key chunks of 16. Scores computed TRANSPOSED (S^T = K . Q^T)
// so the key-axis softmax reduction is 8 in-lane adds + one shfl_xor(16).
// All fragment loads for a WMMA chain are batched first -> progressive waits.
// ---------------------------------------------------------------------------
#define PSTRIDE 18
#define PBUF    (16 * PSTRIDE)

__global__ __launch_bounds__(256) void attn_kernel(
    const float* __restrict__ xq, const float* __restrict__ xk,
    const float* __restrict__ xv, float* __restrict__ out) {
  __shared__ float plds[8 * PBUF];

  const int lane = threadIdx.x & 31;
  const int wid  = threadIdx.x >> 5;
  const int half = lane >> 4;
  const int lm   = lane & 15;

  const int w  = blockIdx.x * 8 + wid;   // 0..4095
  const int qt = w & 63;                 // q tile (S/16 = 64)
  const int h  = (w >> 6) & (NH - 1);
  const int b  = w >> 10;

  const float scale = 0.125f;            // 1/sqrt(64)

  // Q fragments (B operand of S^T GEMM): lane holds Q[q0+lm][4t+2*half+{0,1}]
  const float* qptr = xq + ((size_t)(b * SS + qt * 16 + lm)) * DIMM + h * DH + 2 * half;
  v2f qf[16];
#pragma unroll
  for (int t = 0; t < 16; ++t) qf[t] = *(const v2f*)(qptr + 4 * t);

  v8f o4[4];
#pragma unroll
  for (int j = 0; j < 4; ++j) o4[j] = (v8f){};
  float m_i = -3.0e38f;
  float l_i = 0.0f;

  float* pl = plds + wid * PBUF;

  const float* kptr0 = xk + ((size_t)(b * SS + lm)) * DIMM + h * DH + 2 * half;
  const float* vrow0 = xv + ((size_t)(b * SS)) * DIMM + h * DH;

  for (int kc = 0; kc < SS / 16; ++kc) {
    const int kbase = kc * 16;
    const float* kptr  = kptr0 + (size_t)kbase * DIMM;
    const float* vbase = vrow0 + (size_t)kbase * DIMM;

    // Warm the next chunk's K/V rows while this chunk computes.
    if (kc + 1 < SS / 16) {
      __builtin_prefetch(kptr + (size_t)16 * DIMM, 0, 1);
      __builtin_prefetch(vbase + (size_t)16 * DIMM + (2 * half) * DIMM + lm, 0, 1);
    }

    // ---- S^T[s_k][s_q] = sum_d K[s_k][d] * Q[s_q][d] : batch loads, then MMA
    v2f ka[16];
#pragma unroll
    for (int t = 0; t < 16; ++t) ka[t] = *(const v2f*)(kptr + 4 * t);
    v8f s = {};
#pragma unroll
    for (int t = 0; t < 16; ++t) s = wmma_f32(ka[t], qf[t], s);

    // ---- online softmax over keys (rows of S^T); column s_q = lm per lane
    float cm = -3.0e38f;
#pragma unroll
    for (int r = 0; r < 8; ++r) cm = fmaxf(cm, s[r]);
    cm = fmaxf(cm, __shfl_xor(cm, 16));
    cm *= scale;
    const float nm    = fmaxf(m_i, cm);
    const float alpha = __expf(m_i - nm);
    float p[8];
    float csum = 0.0f;
#pragma unroll
    for (int r = 0; r < 8; ++r) {
      p[r] = __expf(s[r] * scale - nm);
      csum += p[r];
    }
    csum += __shfl_xor(csum, 16);
    l_i = l_i * alpha + csum;
    m_i = nm;

    // ---- V fragments for the whole chunk (issued early, wait amortized)
    v2f vb[16];
#pragma unroll
    for (int ks = 0; ks < 4; ++ks) {
      const int row0 = 4 * ks + 2 * half;
#pragma unroll
      for (int j = 0; j < 4; ++j) {
        vb[ks * 4 + j].x = vbase[(size_t)(row0 + 0) * DIMM + 16 * j + lm];
        vb[ks * 4 + j].y = vbase[(size_t)(row0 + 1) * DIMM + 16 * j + lm];
      }
    }

    // ---- rescale O rows (s_q = r + 8*half) by alpha from lane (r + 8*half)
    float arow[8];
#pragma unroll
    for (int r = 0; r < 8; ++r) arow[r] = __shfl(alpha, r + 8 * half);
#pragma unroll
    for (int j = 0; j < 4; ++j)
#pragma unroll
      for (int r = 0; r < 8; ++r) o4[j][r] *= arow[r];

    // ---- P^T fragment -> P[s_q][s_k] via per-wave LDS (in-order DS ops)
#pragma unroll
    for (int r = 0; r < 8; ++r) pl[lm * PSTRIDE + r + 8 * half] = p[r];
    v2f pa[4];
#pragma unroll
    for (int ks = 0; ks < 4; ++ks)
      pa[ks] = *(const v2f*)(pl + lm * PSTRIDE + 4 * ks + 2 * half);

    // ---- O[s_q][d] += P @ V
#pragma unroll
    for (int ks = 0; ks < 4; ++ks)
#pragma unroll
      for (int j = 0; j < 4; ++j)
        o4[j] = wmma_f32(pa[ks], vb[ks * 4 + j], o4[j]);
  }

  // ---- normalize by 1/l (per row s_q) and store [B,S,H*Dh]
  const float inv = 1.0f / l_i;
  float nrow[8];
#pragma unroll
  for (int r = 0; r < 8; ++r) nrow[r] = __shfl(inv, r + 8 * half);
#pragma unroll
  for (int j = 0; j < 4; ++j) {
#pragma unroll
    for (int r = 0; r < 8; ++r) {
      out[((size_t)(b * SS + qt * 16 + r + 8 * half)) * DIMM + h * DH + 16 * j + lm] =
          o4[j][r] * nrow[r];
    }
  }
}

// ---------------------------------------------------------------------------
// Launch: x, Wq, Wk, Wv, Wo, bo, start_pos  ->  out [B,S,DIM] fp32
// Workspace: xq | xk | xv | attn_out, 16 MB each (64 MB total).
// ---------------------------------------------------------------------------
extern "C" void kernel_launch(void* const* d_in, const int* in_sizes, int n_in,
                              void* d_out, int out_size, void* d_ws, size_t ws_size,
                              hipStream_t stream) {
  const float* x  = (const float*)d_in[0];
  const float* Wq = (const float*)d_in[1];
  const float* Wk = (const float*)d_in[2];
  const float* Wv = (const float*)d_in[3];
  const float* Wo = (const float*)d_in[4];
  const float* bo = (const float*)d_in[5];
  float* out = (float*)d_out;

  const size_t mat = (size_t)MTOT * DIMM;  // 4096*1024
  float* xq = (float*)d_ws;
  float* xk = xq + mat;
  float* xv = xk + mat;
  float* ao = xv + mat;

  // (4096/32)*(1024/64) = 2048 tiles of 32x64, 8 waves/block -> 256 blocks
  const int gemm_blocks = (MTOT / 32) * (DIMM / 64) / 8;

  gemm_nt_wmma<<<gemm_blocks, 256, 0, stream>>>(x, Wq, nullptr, xq, MTOT, DIMM, DIMM);
  gemm_nt_wmma<<<gemm_blocks, 256, 0, stream>>>(x, Wk, nullptr, xk, MTOT, DIMM, DIMM);
  gemm_nt_wmma<<<gemm_blocks, 256, 0, stream>>>(x, Wv, nullptr, xv, MTOT, DIMM, DIMM);

  const int rope_threads = MTOT * (DIMM / 2);
  rope_kernel<<<(rope_threads + 255) / 256, 256, 0, stream>>>(xq, xk);

  // 4 * 16 * 64 = 4096 waves -> 512 blocks of 8 waves
  attn_kernel<<<512, 256, 0, stream>>>(xq, xk, xv, ao);

  gemm_nt_wmma<<<gemm_blocks, 256, 0, stream>>>(ao, Wo, bo, out, MTOT, DIMM, DIMM);
}